// SpaFGAN_83391085019910
// MI455X (gfx1250) — compile-verified
//
#include <hip/hip_runtime.h>

#define NN 50000
#define NN_PAD 50048            // NN rounded up to a multiple of 128 (8 waves x 16 rows)
#define NE 800000
#define EP (NE + NN)            // edges + self loops
#define NEG_SLOPE 0.2f

typedef float v2f __attribute__((ext_vector_type(2)));
typedef float v8f __attribute__((ext_vector_type(8)));

// ---- order-preserving float <-> uint key (for atomicMax-based segment max) ----
__device__ __forceinline__ unsigned fkey(float f) {
    unsigned b = __float_as_uint(f);
    return (b & 0x80000000u) ? ~b : (b | 0x80000000u);
}
__device__ __forceinline__ float funkey(unsigned k) {
    return __uint_as_float((k & 0x80000000u) ? (k & 0x7fffffffu) : ~k);
}
#define NEG_INF_KEY 0x007FFFFFu   // fkey(-inf)

__device__ __forceinline__ float wave_sum(float v) {
#pragma unroll
    for (int o = 16; o > 0; o >>= 1) v += __shfl_xor(v, o, 32);
    return v;
}

__device__ __forceinline__ void edge_nodes(const int* __restrict__ ei, int e, int& s, int& d) {
    if (e < NE) { s = ei[e]; d = ei[NE + e]; }
    else        { s = e - NE; d = e - NE; }   // self loop
}

// ---------------- init workspace ----------------
__global__ void k_init(float* out1, float* out2, unsigned* m1, float* den1,
                       unsigned* m2, float* den2) {
    int gid = blockIdx.x * blockDim.x + threadIdx.x;
    if (gid < NN_PAD * 64) out1[gid] = 0.0f;   // includes GEMM tail padding rows
    if (gid < NN * 32) out2[gid] = 0.0f;
    if (gid < NN * 2) { m1[gid] = NEG_INF_KEY; den1[gid] = 0.0f; }
    if (gid < NN)     { m2[gid] = NEG_INF_KEY; den2[gid] = 0.0f; }
}

// ---------------- layer 1: node transform h1 = x@W1, alpha logits ----------------
// 64 threads per node (2 waves: head0 = ch 0..31, head1 = ch 32..63)
__global__ void k_node1(const float* __restrict__ x, const float* __restrict__ W1,
                        const float* __restrict__ a_src1, const float* __restrict__ a_dst1,
                        float* __restrict__ h1, float* __restrict__ as1, float* __restrict__ ad1) {
    int tid  = threadIdx.x;
    int node = blockIdx.x * 4 + (tid >> 6);
    if (node >= NN) return;
    int c = tid & 63;
    float h = 0.0f;
#pragma unroll
    for (int k = 0; k < 6; k++) h += x[node * 6 + k] * W1[k * 64 + c];
    h1[node * 64 + c] = h;
    int head = c >> 5, cc = c & 31;
    float s = wave_sum(h * a_src1[head * 32 + cc]);
    float d = wave_sum(h * a_dst1[head * 32 + cc]);
    if ((tid & 31) == 0) { as1[node * 2 + head] = s; ad1[node * 2 + head] = d; }
}

// ---------------- layer 1 edge passes ----------------
__global__ void k_edge_max1(const int* __restrict__ ei, const float* __restrict__ as1,
                            const float* __restrict__ ad1, float* __restrict__ e1,
                            unsigned* __restrict__ m1) {
    int gid = blockIdx.x * blockDim.x + threadIdx.x;
    if (gid >= EP * 2) return;
    int e = gid >> 1, head = gid & 1;
    int s, d; edge_nodes(ei, e, s, d);
    float v = as1[s * 2 + head] + ad1[d * 2 + head];
    v = (v > 0.0f) ? v : NEG_SLOPE * v;          // leaky_relu
    e1[gid] = v;
    atomicMax(&m1[d * 2 + head], fkey(v));
}

__global__ void k_edge_exp1(const int* __restrict__ ei, const unsigned* __restrict__ m1,
                            float* __restrict__ e1, float* __restrict__ den1) {
    int gid = blockIdx.x * blockDim.x + threadIdx.x;
    if (gid >= EP * 2) return;
    int e = gid >> 1, head = gid & 1;
    int s, d; edge_nodes(ei, e, s, d);
    float ex = __expf(e1[gid] - funkey(m1[d * 2 + head]));
    e1[gid] = ex;
    atomicAdd(&den1[d * 2 + head], ex);
}

__global__ void k_edge_agg1(const int* __restrict__ ei, const float* __restrict__ e1,
                            const float* __restrict__ den1, const float* __restrict__ h1,
                            float* __restrict__ out1) {
    int gid = blockIdx.x * blockDim.x + threadIdx.x;   // EP*64 = 54.4M < 2^31
    if (gid >= EP * 64) return;
    int e = gid >> 6, c = gid & 63, head = c >> 5;
    int s, d; edge_nodes(ei, e, s, d);
    float w = e1[e * 2 + head] / den1[d * 2 + head];
    atomicAdd(&out1[d * 64 + c], w * h1[s * 64 + c]);
}

// ---------------- layer 2 GEMM via V_WMMA_F32_16X16X4_F32 ----------------
// h2[NN_PAD,32] = relu(out1 + b1) @ W2[64,32].  One wave -> 16 nodes x 32 cols.
// Node dim padded to NN_PAD: no guards, no divergence, EXEC all-1s throughout.
__global__ void k_gemm2(const float* __restrict__ out1, const float* __restrict__ pb1,
                        const float* __restrict__ W2, float* __restrict__ h2) {
    __shared__ float sW[64 * 32];                // stage W2 (8 KB) in LDS
    int tid = threadIdx.x;
#pragma unroll
    for (int i = 0; i < 8; i++) sW[tid + i * 256] = W2[tid + i * 256];
    __syncthreads();

    int lane  = tid & 31;
    int wave  = tid >> 5;
    int node0 = (blockIdx.x * 8 + wave) * 16;    // < NN_PAD only for blockIdx.x < 391
    int m     = lane & 15;                       // A row / B,D col within tile
    int half  = lane >> 4;                       // 0 or 1
    int kb    = half << 1;                       // K pair base: 0 or 2
    const float* arow = out1 + (size_t)(node0 + m) * 64;

    v8f acc0 = {0.f,0.f,0.f,0.f,0.f,0.f,0.f,0.f};
    v8f acc1 = {0.f,0.f,0.f,0.f,0.f,0.f,0.f,0.f};
#pragma unroll
    for (int k0 = 0; k0 < 64; k0 += 4) {
        int k = k0 + kb;                         // even -> 8B-aligned pair
        v2f a    = *(const v2f*)(arow + k);      // global_load_b64
        v2f bias = *(const v2f*)(pb1 + k);
        a.x = fmaxf(a.x + bias.x, 0.0f);         // fused bias + ReLU
        a.y = fmaxf(a.y + bias.y, 0.0f);
        v2f bA, bB;
        bA.x = sW[k * 32 + m];        bA.y = sW[(k + 1) * 32 + m];        // cols 0..15
        bB.x = sW[k * 32 + 16 + m];   bB.y = sW[(k + 1) * 32 + 16 + m];   // cols 16..31
        acc0 = __builtin_amdgcn_wmma_f32_16x16x4_f32(false, a, false, bA, (short)0, acc0, false, false);
        acc1 = __builtin_amdgcn_wmma_f32_16x16x4_f32(false, a, false, bB, (short)0, acc1, false, false);
    }
    int rbase = node0 + (half << 3);             // D: VGPR r -> row r (+8 for lanes 16..31)
#pragma unroll
    for (int r = 0; r < 8; r++) {
        h2[(rbase + r) * 32 + m]      = acc0[r]; // padded rows written, never read
        h2[(rbase + r) * 32 + 16 + m] = acc1[r];
    }
}

// ---------------- layer 2 alpha logits (1 head) ----------------
__global__ void k_alpha2(const float* __restrict__ h2, const float* __restrict__ a_src2,
                         const float* __restrict__ a_dst2, float* __restrict__ as2,
                         float* __restrict__ ad2) {
    int lane = threadIdx.x & 31;
    int node = blockIdx.x * 8 + (threadIdx.x >> 5);
    if (node >= NN) return;
    float h = h2[node * 32 + lane];
    float s = wave_sum(h * a_src2[lane]);
    float d = wave_sum(h * a_dst2[lane]);
    if (lane == 0) { as2[node] = s; ad2[node] = d; }
}

// ---------------- layer 2 edge passes ----------------
__global__ void k_edge_max2(const int* __restrict__ ei, const float* __restrict__ as2,
                            const float* __restrict__ ad2, float* __restrict__ e2,
                            unsigned* __restrict__ m2) {
    int gid = blockIdx.x * blockDim.x + threadIdx.x;
    if (gid >= EP) return;
    int s, d; edge_nodes(ei, gid, s, d);
    float v = as2[s] + ad2[d];
    v = (v > 0.0f) ? v : NEG_SLOPE * v;
    e2[gid] = v;
    atomicMax(&m2[d], fkey(v));
}

__global__ void k_edge_exp2(const int* __restrict__ ei, const unsigned* __restrict__ m2,
                            float* __restrict__ e2, float* __restrict__ den2) {
    int gid = blockIdx.x * blockDim.x + threadIdx.x;
    if (gid >= EP) return;
    int s, d; edge_nodes(ei, gid, s, d);
    float ex = __expf(e2[gid] - funkey(m2[d]));
    e2[gid] = ex;
    atomicAdd(&den2[d], ex);
}

__global__ void k_edge_agg2(const int* __restrict__ ei, const float* __restrict__ e2,
                            const float* __restrict__ den2, const float* __restrict__ h2,
                            float* __restrict__ out2) {
    int gid = blockIdx.x * blockDim.x + threadIdx.x;   // EP*32 = 27.2M
    if (gid >= EP * 32) return;
    int e = gid >> 5, c = gid & 31;
    int s, d; edge_nodes(ei, e, s, d);
    float w = e2[e] / den2[d];
    atomicAdd(&out2[d * 32 + c], w * h2[s * 32 + c]);
}

// ---------------- final: sigmoid(relu(out2 + b2) @ Wout + bout) ----------------
__global__ void k_final(const float* __restrict__ out2, const float* __restrict__ b2,
                        const float* __restrict__ Wout, const float* __restrict__ bout,
                        float* __restrict__ out) {
    int lane = threadIdx.x & 31;
    int node = blockIdx.x * 8 + (threadIdx.x >> 5);
    if (node >= NN) return;
    float v = fmaxf(out2[node * 32 + lane] + b2[lane], 0.0f) * Wout[lane];
    v = wave_sum(v);
    if (lane == 0) {
        float z = v + bout[0];
        out[node] = 1.0f / (1.0f + __expf(-z));
    }
}

extern "C" void kernel_launch(void* const* d_in, const int* in_sizes, int n_in,
                              void* d_out, int out_size, void* d_ws, size_t ws_size,
                              hipStream_t stream) {
    const float* x      = (const float*)d_in[0];
    const int*   ei     = (const int*)d_in[1];
    const float* W1     = (const float*)d_in[2];
    const float* a_src1 = (const float*)d_in[3];
    const float* a_dst1 = (const float*)d_in[4];
    const float* b1     = (const float*)d_in[5];
    const float* W2     = (const float*)d_in[6];
    const float* a_src2 = (const float*)d_in[7];
    const float* a_dst2 = (const float*)d_in[8];
    const float* b2     = (const float*)d_in[9];
    const float* Wout   = (const float*)d_in[10];
    const float* bout   = (const float*)d_in[11];
    float* out = (float*)d_out;

    char* base = (char*)d_ws;
    size_t off = 0;
    auto alloc = [&](size_t bytes) -> char* {
        size_t cur = (off + 255) & ~(size_t)255;
        off = cur + bytes;
        return base + cur;
    };
    float*    h1   = (float*)alloc((size_t)NN * 64 * 4);
    float*    out1 = (float*)alloc((size_t)NN_PAD * 64 * 4);   // GEMM A (padded)
    float*    h2   = (float*)alloc((size_t)NN_PAD * 32 * 4);   // GEMM D (padded)
    float*    out2 = (float*)alloc((size_t)NN * 32 * 4);
    float*    e1   = (float*)alloc((size_t)EP * 2 * 4);
    float*    e2   = (float*)alloc((size_t)EP * 4);
    float*    as1  = (float*)alloc((size_t)NN * 2 * 4);
    float*    ad1  = (float*)alloc((size_t)NN * 2 * 4);
    float*    den1 = (float*)alloc((size_t)NN * 2 * 4);
    unsigned* m1   = (unsigned*)alloc((size_t)NN * 2 * 4);
    float*    as2  = (float*)alloc((size_t)NN * 4);
    float*    ad2  = (float*)alloc((size_t)NN * 4);
    float*    den2 = (float*)alloc((size_t)NN * 4);
    unsigned* m2   = (unsigned*)alloc((size_t)NN * 4);

    const int T = 256;
    auto gb = [](long long n, int t) { return dim3((unsigned)((n + t - 1) / t)); };

    k_init<<<gb((long long)NN_PAD * 64, T), T, 0, stream>>>(out1, out2, m1, den1, m2, den2);
    k_node1<<<gb((long long)NN * 64, T), T, 0, stream>>>(x, W1, a_src1, a_dst1, h1, as1, ad1);
    k_edge_max1<<<gb((long long)EP * 2, T), T, 0, stream>>>(ei, as1, ad1, e1, m1);
    k_edge_exp1<<<gb((long long)EP * 2, T), T, 0, stream>>>(ei, m1, e1, den1);
    k_edge_agg1<<<gb((long long)EP * 64, T), T, 0, stream>>>(ei, e1, den1, h1, out1);
    k_gemm2<<<gb((long long)NN_PAD * 16, T), T, 0, stream>>>(out1, b1, W2, h2);  // 3128 blocks; only the first 391 cover NN_PAD
    k_alpha2<<<gb((long long)NN * 32, T), T, 0, stream>>>(h2, a_src2, a_dst2, as2, ad2);
    k_edge_max2<<<gb((long long)EP, T), T, 0, stream>>>(ei, as2, ad2, e2, m2);
    k_edge_exp2<<<gb((long long)EP, T), T, 0, stream>>>(ei, m2, e2, den2);
    k_edge_agg2<<<gb((long long)EP * 32, T), T, 0, stream>>>(ei, e2, den2, h2, out2);
    k_final<<<gb((long long)NN * 32, T), T, 0, stream>>>(out2, b2, Wout, bout, out);
}